// pos_H_MT_version_1_18622978195605
// MI455X (gfx1250) — compile-verified
//
#include <hip/hip_runtime.h>

// Multi-threshold spiking step over [T,B,C,H,W] f32, streaming elementwise along
// T-chains. Memory-bound: ~410 MB traffic -> ~18 us floor at 23.3 TB/s. No matmul
// structure, so no WMMA; optimal path is wave32 + b128 non-temporal streaming.
//
// Ladder identity: greedy binary ladder over levels lam*2^(NT-1)..lam*2^0 equals
//   q = clamp(floor(mem/lam), 0, 2^NT-1);  out = q*lam;  mem -= out;
// For power-of-two lam (setup: lam=1) every step of this closed form is exact and
// bit-identical to the reference's sequential subtraction. Non-pow2 lam falls back
// to the literal ladder.

typedef float v4f __attribute__((ext_vector_type(4)));

// ---- exact literal ladder (general lambda fallback) ----
__device__ __forceinline__ float mt_ladder(float& m, float thr0, int nt) {
    float o = 0.0f;
    float thr = thr0;
    for (int i = 0; i < nt; ++i) {
        float s = (m >= thr) ? thr : 0.0f;
        o += s;
        m -= s;
        thr *= 0.5f;
    }
    return o;
}

__device__ __forceinline__ v4f mt_step4_ladder(v4f& mem, v4f xi, float thr0, int nt) {
    v4f o;
#pragma unroll
    for (int k = 0; k < 4; ++k) {
        float m = mem[k] + xi[k];
        o[k] = mt_ladder(m, thr0, nt);
        mem[k] = m;
    }
    return o;
}

// ---- closed form (power-of-two lambda): ~7 VALU ops/element-step, loop-free ----
__device__ __forceinline__ v4f mt_step4_closed(v4f& mem, v4f xi,
                                               float lam, float inv_lam, float qmax) {
    v4f o;
#pragma unroll
    for (int k = 0; k < 4; ++k) {
        float m = mem[k] + xi[k];                     // v_add
        float q = floorf(m * inv_lam);                // v_mul + v_floor (exact: pow2 scale)
        q = fminf(fmaxf(q, 0.0f), qmax);              // v_max_num + v_min_num
        float s = q * lam;                            // v_mul (exact)
        o[k] = s;
        mem[k] = m - s;                               // v_sub (Sterbenz-exact)
    }
    return o;
}

extern "C" __global__ __launch_bounds__(256) void snn_mt_kernel(
    const float* __restrict__ x,
    const float* __restrict__ lam_p,
    const int*   __restrict__ t_p,
    const int*   __restrict__ nt_p,
    float*       __restrict__ out,
    long long total)
{
    const int   T   = t_p[0];                 // scalar params live on device
    const int   NT  = nt_p[0];
    const float lam = lam_p[0];
    const long long N = total / (long long)T; // elements per timestep (contiguous)

    const float thr0 = lam * (float)(1u << (NT - 1));
    const float m0   = 0.5f * lam;

    // Wave-uniform check: lambda a normal positive power of two?
    union { float f; unsigned u; } lu; lu.f = lam;
    const unsigned expo = (lu.u >> 23) & 0xFFu;
    const bool pow2_lam = ((lu.u & 0x80000000u) == 0u) &&
                          ((lu.u & 0x007FFFFFu) == 0u) &&
                          (expo >= 1u) && (expo <= 254u);
    const float inv_lam = 1.0f / lam;                       // exact when pow2
    const float qmax    = (float)((1u << NT) - 1u);

    const long long tid    = (long long)blockIdx.x * blockDim.x + threadIdx.x;
    const long long stride = (long long)gridDim.x * blockDim.x;

    // Vector path only when each timestep block stays 16B-aligned.
    const long long nslots = (N & 3LL) ? 0LL : (N >> 2);

    if (pow2_lam) {
        // ---------------- fast closed-form path ----------------
        for (long long s = tid; s < nslots; s += stride) {
            const long long base = s << 2;
            v4f mem = { m0, m0, m0, m0 };

            if (T == 4) {
                // Issue all 4 b128 NT loads up front; LOADcnt overlaps the
                // serial membrane chain. Then 4 NT stores.
                const v4f x0 = __builtin_nontemporal_load((const v4f*)(x +         base));
                const v4f x1 = __builtin_nontemporal_load((const v4f*)(x +   N   + base));
                const v4f x2 = __builtin_nontemporal_load((const v4f*)(x + 2LL*N + base));
                const v4f x3 = __builtin_nontemporal_load((const v4f*)(x + 3LL*N + base));

                v4f o0 = mt_step4_closed(mem, x0, lam, inv_lam, qmax);
                v4f o1 = mt_step4_closed(mem, x1, lam, inv_lam, qmax);
                v4f o2 = mt_step4_closed(mem, x2, lam, inv_lam, qmax);
                v4f o3 = mt_step4_closed(mem, x3, lam, inv_lam, qmax);

                __builtin_nontemporal_store(o0, (v4f*)(out +         base));
                __builtin_nontemporal_store(o1, (v4f*)(out +   N   + base));
                __builtin_nontemporal_store(o2, (v4f*)(out + 2LL*N + base));
                __builtin_nontemporal_store(o3, (v4f*)(out + 3LL*N + base));
            } else {
                v4f xv = __builtin_nontemporal_load((const v4f*)(x + base));
                for (int t = 0; t < T; ++t) {
                    v4f xn = { 0.f, 0.f, 0.f, 0.f };
                    if (t + 1 < T)
                        xn = __builtin_nontemporal_load(
                            (const v4f*)(x + (long long)(t + 1) * N + base));
                    v4f o = mt_step4_closed(mem, xv, lam, inv_lam, qmax);
                    __builtin_nontemporal_store(o, (v4f*)(out + (long long)t * N + base));
                    xv = xn;
                }
            }
        }
        // scalar tail
        const long long tail0 = nslots << 2;
        for (long long e = tail0 + tid; e < N; e += stride) {
            float m = m0;
            for (int t = 0; t < T; ++t) {
                m += x[(long long)t * N + e];
                float q = fminf(fmaxf(floorf(m * inv_lam), 0.0f), qmax);
                float o = q * lam;
                m -= o;
                out[(long long)t * N + e] = o;
            }
        }
    } else {
        // ---------------- exact ladder fallback (any lambda) ----------------
        for (long long s = tid; s < nslots; s += stride) {
            const long long base = s << 2;
            v4f mem = { m0, m0, m0, m0 };

            if (T == 4) {
                const v4f x0 = __builtin_nontemporal_load((const v4f*)(x +         base));
                const v4f x1 = __builtin_nontemporal_load((const v4f*)(x +   N   + base));
                const v4f x2 = __builtin_nontemporal_load((const v4f*)(x + 2LL*N + base));
                const v4f x3 = __builtin_nontemporal_load((const v4f*)(x + 3LL*N + base));

                v4f o0 = mt_step4_ladder(mem, x0, thr0, NT);
                v4f o1 = mt_step4_ladder(mem, x1, thr0, NT);
                v4f o2 = mt_step4_ladder(mem, x2, thr0, NT);
                v4f o3 = mt_step4_ladder(mem, x3, thr0, NT);

                __builtin_nontemporal_store(o0, (v4f*)(out +         base));
                __builtin_nontemporal_store(o1, (v4f*)(out +   N   + base));
                __builtin_nontemporal_store(o2, (v4f*)(out + 2LL*N + base));
                __builtin_nontemporal_store(o3, (v4f*)(out + 3LL*N + base));
            } else {
                v4f xv = __builtin_nontemporal_load((const v4f*)(x + base));
                for (int t = 0; t < T; ++t) {
                    v4f xn = { 0.f, 0.f, 0.f, 0.f };
                    if (t + 1 < T)
                        xn = __builtin_nontemporal_load(
                            (const v4f*)(x + (long long)(t + 1) * N + base));
                    v4f o = mt_step4_ladder(mem, xv, thr0, NT);
                    __builtin_nontemporal_store(o, (v4f*)(out + (long long)t * N + base));
                    xv = xn;
                }
            }
        }
        const long long tail0 = nslots << 2;
        for (long long e = tail0 + tid; e < N; e += stride) {
            float m = m0;
            for (int t = 0; t < T; ++t) {
                m += x[(long long)t * N + e];
                out[(long long)t * N + e] = mt_ladder(m, thr0, NT);
            }
        }
    }
}

extern "C" void kernel_launch(void* const* d_in, const int* in_sizes, int n_in,
                              void* d_out, int out_size, void* d_ws, size_t ws_size,
                              hipStream_t stream) {
    const float* x   = (const float*)d_in[0];   // [T*B, C, H, W] fp32
    const float* lam = (const float*)d_in[1];   // scalar lambda_min
    const int*   T   = (const int*)d_in[2];     // scalar T (device)
    const int*   NT  = (const int*)d_in[3];     // scalar num_thresholds (device)
    float*       out = (float*)d_out;

    const long long total = (long long)in_sizes[0];

    // One thread per 4 elements per chain, sized for the nominal T=4 case;
    // the grid-stride loop keeps it correct for any device-side T.
    const int block = 256;  // 8 wave32 waves per block
    long long threads_needed = (total + 15) / 16;
    long long blocks = (threads_needed + block - 1) / block;
    if (blocks < 1) blocks = 1;
    if (blocks > 0x7FFFFFFFLL) blocks = 0x7FFFFFFFLL;

    snn_mt_kernel<<<dim3((unsigned)blocks), dim3(block), 0, stream>>>(
        x, lam, T, NT, out, total);
}